// ScaleDotProductAttention_76433237999811
// MI455X (gfx1250) — compile-verified
//
#include <hip/hip_runtime.h>
#include <stdint.h>

// Causal attention fwd, B=4, H=16, S=2048, D=64 (fp32), returns (out, score).
// Memory-bound on the 1.07GB score materialization -> native fp32 WMMA path
// (V_WMMA_F32_16X16X4_F32), TDM async V-tile staging, ds_swizzle reductions.

#define S_LEN 2048
#define D_DIM 64
#define NEGV  -65500.0f

typedef __attribute__((ext_vector_type(2))) float v2f;
typedef __attribute__((ext_vector_type(8))) float v8f;

// ---------------- WMMA wrapper ----------------
__device__ __forceinline__ v8f wmma4(v2f a, v2f b, v8f c) {
    return __builtin_amdgcn_wmma_f32_16x16x4_f32(false, a, false, b,
                                                 (short)0, c, false, false);
}

// ---------------- cross-lane XOR reductions (within 16-lane half) ----------
#if __has_builtin(__builtin_amdgcn_ds_swizzle)
template <int M>
__device__ __forceinline__ float xmax(float x) {
    int i = __builtin_amdgcn_ds_swizzle(__float_as_int(x), (M << 10) | 0x1f);
    return fmaxf(x, __int_as_float(i));
}
template <int M>
__device__ __forceinline__ float xadd(float x) {
    int i = __builtin_amdgcn_ds_swizzle(__float_as_int(x), (M << 10) | 0x1f);
    return x + __int_as_float(i);
}
#else
template <int M>
__device__ __forceinline__ float xmax(float x) { return fmaxf(x, __shfl_xor(x, M, 32)); }
template <int M>
__device__ __forceinline__ float xadd(float x) { return x + __shfl_xor(x, M, 32); }
#endif

// ---------------- Tensor Data Mover (async V-tile -> LDS) ------------------
#if __has_builtin(__builtin_amdgcn_tensor_load_to_lds) && \
    __has_builtin(__builtin_amdgcn_s_wait_tensorcnt)
#define USE_TDM 1
typedef __attribute__((ext_vector_type(4))) unsigned int v4u;
typedef __attribute__((ext_vector_type(4))) int         v4i_t;
typedef __attribute__((ext_vector_type(8))) int         v8i_t;

// DMA one 16x64 fp32 tile (4KB, contiguous rows, stride 64) into LDS.
// clang-23 6-arg form: (v4u g0, v8i g1, v4i g2, v4i g3, v8i g4, i32 cpol)
__device__ __forceinline__ void tdm_load_v_tile(uint32_t lds_off, const float* gsrc) {
    uint64_t ga = (uint64_t)(uintptr_t)gsrc;
    v4u g0;
    g0.x = 1u;                                            // count=1 (valid D#)
    g0.y = lds_off;                                       // lds_addr
    g0.z = (uint32_t)ga;                                  // global_addr[31:0]
    g0.w = (uint32_t)((ga >> 32) & 0x01FFFFFFu) | (2u << 30); // addr[56:32] | type=2
    v8i_t g1;
    g1[0] = (int)(2u << 16);       // data_size=4B; wg_mask=0
    g1[1] = (int)(64u << 16);      // tensor_dim0 = 64 (low16 in [31:16])
    g1[2] = (int)(2048u << 16);    // dim0 hi=0; tensor_dim1 = 2048 (low16)
    g1[3] = (int)(64u << 16);      // dim1 hi=0; tile_dim0 = 64
    g1[4] = 16;                    // tile_dim1 = 16; tile_dim2 = 0
    g1[5] = 64;                    // tensor_dim0_stride = 64 (low 32)
    g1[6] = 0;                     // stride0 hi / stride1 lo
    g1[7] = 0;                     // stride1 hi
    v4i_t g2 = {0, 0, 0, 0};       // unused (2D tensor)
    v4i_t g3 = {0, 0, 0, 0};       // unused (2D tensor)
    v8i_t g4;
    g4[0] = 0; g4[1] = 0; g4[2] = 0; g4[3] = 0;
    g4[4] = 0; g4[5] = 0; g4[6] = 0; g4[7] = 0;
    __builtin_amdgcn_tensor_load_to_lds(g0, g1, g2, g3, g4, 0);
}
#endif

__global__ __launch_bounds__(256) void
sdpa_fp32_wmma_kernel(const float* __restrict__ q,
                      const float* __restrict__ k,
                      const float* __restrict__ v,
                      float* __restrict__ out,
                      float* __restrict__ score)
{
    __shared__ float lds_p[8][16 * 16];          // P staging (C->A relayout)
#ifdef USE_TDM
    __shared__ float lds_v[8][2][16 * D_DIM];    // double-buffered V tiles
#endif

    const int lane = threadIdx.x & 31;
    const int wave = threadIdx.x >> 5;
    const int hi   = lane >> 4;
    const int ln   = lane & 15;

    const int idx = blockIdx.x;            // [0, B*H * S/128)
    const int bh  = idx >> 4;
    const int rb  = idx & 15;
    const int qr  = rb * 128 + wave * 16;  // first query row of this wave
    const int qt  = qr >> 4;               // diagonal key-tile index

    const float* qbase = q + ((size_t)bh * S_LEN + qr) * D_DIM;
    const float* kbase = k + (size_t)bh * S_LEN * D_DIM;
    const float* vbase = v + (size_t)bh * S_LEN * D_DIM;
    float*       obase = out   + ((size_t)bh * S_LEN + qr) * D_DIM;
    float*       sbase = score + ((size_t)bh * S_LEN + qr) * (size_t)S_LEN;

    // ---- preload Q tile as 16 WMMA-A f32 fragments ----
    v2f qa[16];
#pragma unroll
    for (int j = 0; j < 16; ++j)
        qa[j] = *(const v2f*)(qbase + ln * D_DIM + 4 * j + 2 * hi);

    const float scale = 0.125f;  // 1/sqrt(64)

    // 16 chained k=4 WMMAs for one 16x16 score tile, dual accumulators for ILP
    auto qk16 = [&](int kt, v8f& cacc) {
        const float* kb = kbase + (size_t)(kt * 16 + ln) * D_DIM + 2 * hi;
        v8f c0 = {}, c1 = {};
#pragma unroll
        for (int j = 0; j < 8; ++j) {
            v2f b0 = *(const v2f*)(kb + 4 * j);
            c0 = wmma4(qa[j], b0, c0);
            v2f b1 = *(const v2f*)(kb + 4 * (j + 8));
            c1 = wmma4(qa[j + 8], b1, c1);
        }
        cacc = c0 + c1;
    };

    float m[8], l[8];
#pragma unroll
    for (int r = 0; r < 8; ++r) { m[r] = -3.0e38f; l[r] = 0.0f; }

    auto update_stats = [&](const v8f& c, int kt, bool masked) {
#pragma unroll
        for (int r = 0; r < 8; ++r) {
            float s = c[r] * scale;
            if (masked) {
                const int rowg = qr + r + 8 * hi;
                const int colg = kt * 16 + ln;
                if (colg > rowg) s = NEGV;
            }
            const float rm = xmax<8>(xmax<4>(xmax<2>(xmax<1>(s))));
            const float nm = fmaxf(m[r], rm);
            const float e  = __expf(s - nm);
            const float rs = xadd<8>(xadd<4>(xadd<2>(xadd<1>(e))));
            l[r] = l[r] * __expf(m[r] - nm) + rs;
            m[r] = nm;
        }
    };

    // ================= pass 1: row max & sum(exp) =================
    for (int kt = 0; kt < qt; ++kt) {   // off-diagonal: no mask needed
        __builtin_prefetch(kbase + (size_t)((kt + 1) * 16 + ln) * D_DIM, 0, 1);
        v8f c;
        qk16(kt, c);
        update_stats(c, kt, false);
    }
    {                                    // diagonal tile: causal mask
        v8f c;
        qk16(qt, c);
        update_stats(c, qt, true);
    }

    float invl[8];
#pragma unroll
    for (int r = 0; r < 8; ++r) invl[r] = 1.0f / l[r];

    // ================= pass 2: P store + O = P*V =================
    v8f o[4] = {};
    float* myp = &lds_p[wave][0];

#ifdef USE_TDM
    float*   vtile[2] = { &lds_v[wave][0][0], &lds_v[wave][1][0] };
    uint32_t voff[2]  = { (uint32_t)(uintptr_t)vtile[0],
                          (uint32_t)(uintptr_t)vtile[1] };
    tdm_load_v_tile(voff[0], vbase);     // prime buffer 0 with V tile kt=0
#endif

    auto pv_tile = [&](int kt, bool masked) {
        v8f c;
        qk16(kt, c);
        // probabilities -> global score + LDS staging (C-layout)
#pragma unroll
        for (int r = 0; r < 8; ++r) {
            const int row  = r + 8 * hi;
            const int colg = kt * 16 + ln;
            float s = c[r] * scale;
            if (masked && colg > qr + row) s = NEGV;
            const float p = __expf(s - m[r]) * invl[r];
            sbase[(size_t)row * S_LEN + colg] = p;
            myp[row * 16 + ln] = p;
        }
        // cross-lane LDS handoff within this wave
        asm volatile("s_wait_dscnt 0" ::: "memory");
        __builtin_amdgcn_wave_barrier();

#ifdef USE_TDM
        // prior DS reads of the other V buffer are now drained -> safe to
        // overwrite it with the next tile's DMA while we do P*V on this one.
        if (kt < qt)
            tdm_load_v_tile(voff[(kt + 1) & 1], vbase + (size_t)(kt + 1) * 16 * D_DIM);
#endif
        // re-read P as WMMA-A fragments
        v2f pa[4];
#pragma unroll
        for (int j = 0; j < 4; ++j)
            pa[j] = *(const v2f*)(myp + ln * 16 + 4 * j + 2 * hi);

#ifdef USE_TDM
        if (kt < qt) __builtin_amdgcn_s_wait_tensorcnt(1);  // this tile's DMA done
        else         __builtin_amdgcn_s_wait_tensorcnt(0);
        asm volatile("" ::: "memory");
        const float* vt = vtile[kt & 1];
#endif
        // O[n-chunk] += P * V
#pragma unroll
        for (int n = 0; n < 4; ++n) {
#pragma unroll
            for (int j = 0; j < 4; ++j) {
                v2f b;
#ifdef USE_TDM
                b.x = vt[(4 * j + 2 * hi) * D_DIM + n * 16 + ln];
                b.y = vt[(4 * j + 2 * hi + 1) * D_DIM + n * 16 + ln];
#else
                const float* vrow = vbase + (size_t)(kt * 16 + 4 * j + 2 * hi) * D_DIM + n * 16 + ln;
                b.x = vrow[0];
                b.y = vrow[D_DIM];
#endif
                o[n] = wmma4(pa[j], b, o[n]);
            }
        }
    };

    for (int kt = 0; kt < qt; ++kt) {
        __builtin_prefetch(kbase + (size_t)((kt + 1) * 16 + ln) * D_DIM, 0, 1);
        pv_tile(kt, false);
    }
    pv_tile(qt, true);

    // ---- zero-fill the fully-masked upper-triangle score columns ----
    {
        const int cstart = (qt + 1) * 16;
        const float4 z = make_float4(0.f, 0.f, 0.f, 0.f);
        for (int row = 0; row < 16; ++row) {
            float* srow = sbase + (size_t)row * S_LEN;
            for (int c0 = cstart + lane * 4; c0 < S_LEN; c0 += 128)
                *(float4*)(srow + c0) = z;
        }
    }

    // ---- write O tile ----
#pragma unroll
    for (int n = 0; n < 4; ++n)
#pragma unroll
        for (int r = 0; r < 8; ++r)
            obase[(size_t)(r + 8 * hi) * D_DIM + n * 16 + ln] = o[n][r];
}

extern "C" void kernel_launch(void* const* d_in, const int* in_sizes, int n_in,
                              void* d_out, int out_size, void* d_ws, size_t ws_size,
                              hipStream_t stream) {
    (void)in_sizes; (void)n_in; (void)d_ws; (void)ws_size; (void)out_size;
    const float* q = (const float*)d_in[0];
    const float* k = (const float*)d_in[1];
    const float* v = (const float*)d_in[2];
    // d_in[3] is the causal mask; computed analytically in-kernel.

    const int B = 4, H = 16;
    float* out   = (float*)d_out;                          // [B,H,S,D]
    float* score = out + (size_t)B * H * S_LEN * D_DIM;    // [B,H,S,S]

    const int blocks = B * H * (S_LEN / 128);              // 1024
    sdpa_fp32_wmma_kernel<<<blocks, 256, 0, stream>>>(q, k, v, out, score);
}